// BLLoss_66494683676972
// MI455X (gfx1250) — compile-verified
//
#include <hip/hip_runtime.h>
#include <math.h>

// ---------------------------------------------------------------------------
// NT-Xent (SimCLR) loss for B=2048, D=512, tau=0.5.
// sim = rep @ rep^T via V_WMMA_F32_16X16X4_F32. Each wave computes a 64x64
// macro-tile (4x4 grid of 16x16 WMMA tiles, 16 v8f accumulators) so each
// K-step issues 8 b64 loads for 16 WMMAs (0.5 loads/WMMA). Upper-triangle
// macros only (symmetry), accumulators scaled by reciprocal row norms,
// exp via v_exp_f32, fp64 atomic accumulation.
// ---------------------------------------------------------------------------

typedef float v2f __attribute__((ext_vector_type(2)));
typedef float v8f __attribute__((ext_vector_type(8)));

#define DIM    512
#define NROWS  8192
#define HALF   4096          // rows per input tensor (2B)
#define NM     128           // 64-row macro tiles per matrix dim
#define MT     8256          // NM*(NM+1)/2 upper-triangle macro tiles

// ---- Kernel 1: reciprocal row norms ---------------------------------------
__global__ __launch_bounds__(256)
void ntx_rownorm_kernel(const float* __restrict__ ei,
                        const float* __restrict__ ej,
                        float* __restrict__ inv)
{
    const int row = blockIdx.x;
    const float* src = (row < HALF) ? (ei + (size_t)row * DIM)
                                    : (ej + (size_t)(row - HALF) * DIM);
    const int tid = threadIdx.x;                  // 256 threads, 2 elems each
    float x0 = src[tid];
    float x1 = src[tid + 256];
    __shared__ float red[256];
    red[tid] = x0 * x0 + x1 * x1;
    __syncthreads();
    for (int off = 128; off > 0; off >>= 1) {
        if (tid < off) red[tid] += red[tid + off];
        __syncthreads();
    }
    if (tid == 0) {
        float n = sqrtf(red[0]);
        inv[row] = 1.0f / fmaxf(n, 1e-12f);      // matches F.normalize eps
    }
}

// ---- Kernel 2: 64x64 macro-tiled WMMA GEMM + exp-reduce -------------------
__global__ __launch_bounds__(256)
void ntx_sim_kernel(const float* __restrict__ ei,
                    const float* __restrict__ ej,
                    const float* __restrict__ inv,
                    double* __restrict__ acc)     // acc[0]=total offdiag, acc[1]=nominator
{
    const int lane   = threadIdx.x & 31;
    const int waveId = threadIdx.x >> 5;
    const int wid    = blockIdx.x * 8 + waveId;   // 8 waves per block (wave32)
    const int nwaves = gridDim.x * 8;

    const int n = lane & 15;                      // tile row (A) / tile col (B,C)
    const int h = lane >> 4;                      // half-wave selects K pair

    float tot = 0.0f;                             // weighted sum exp(2s), i!=j
    float nom = 0.0f;                             // positives-diagonals part

    for (int t = wid; t < MT; t += nwaves) {
        // map linear triangle index -> (ti, tj), tj >= ti, over NM=128 rows.
        // S(x) = x*(2*NM+1-x)/2 macros precede row x.
        long long tt = t;
        int ti = (int)((257.0 - sqrt(257.0 * 257.0 - 8.0 * (double)tt)) * 0.5);
        if (ti < 0) ti = 0;
        if (ti > NM - 1) ti = NM - 1;
        while (ti > 0 && (long long)ti * (257 - ti) / 2 > tt) --ti;
        while ((long long)(ti + 1) * (257 - (ti + 1)) / 2 <= tt) ++ti;
        const int tj = ti + (int)(tt - (long long)ti * (257 - ti) / 2);

        const int mi0 = ti << 6;                  // first row of macro (i)
        const int mj0 = tj << 6;                  // first row of macro (j)

        // each 64-row macro lies entirely inside one input tensor (4096%64==0)
        const float* Ab = (mi0 < HALF) ? (ei + (size_t)mi0 * DIM)
                                       : (ej + (size_t)(mi0 - HALF) * DIM);
        const float* Bb = (mj0 < HALF) ? (ei + (size_t)mj0 * DIM)
                                       : (ej + (size_t)(mj0 - HALF) * DIM);

        // A 16x4 fp32: lane -> {A[m][2h], A[m][2h+1]},   m = lane&15
        // B 4x16 fp32: lane -> {B[2h][n], B[2h+1][n]} = row_j[n][2h..2h+1]
        const float* ap = Ab + (size_t)n * DIM + 2 * h;
        const float* bp = Bb + (size_t)n * DIM + 2 * h;

        v8f c[4][4];
        #pragma unroll
        for (int si = 0; si < 4; ++si)
            #pragma unroll
            for (int sj = 0; sj < 4; ++sj)
                c[si][sj] = (v8f){0.f, 0.f, 0.f, 0.f, 0.f, 0.f, 0.f, 0.f};

        #pragma unroll 2
        for (int k = 0; k < DIM; k += 4) {
            v2f a[4], b[4];
            #pragma unroll
            for (int s = 0; s < 4; ++s) {
                // sub-tile row offsets (s*16 rows = s*32KB) fold into ioffset
                a[s] = *(const v2f*)(ap + (size_t)(s * 16) * DIM + k);
                b[s] = *(const v2f*)(bp + (size_t)(s * 16) * DIM + k);
            }
            #pragma unroll
            for (int si = 0; si < 4; ++si)
                #pragma unroll
                for (int sj = 0; sj < 4; ++sj)
                    // 8 args: (neg_a, A, neg_b, B, c_mod, C, reuse_a, reuse_b)
                    c[si][sj] = __builtin_amdgcn_wmma_f32_16x16x4_f32(
                        false, a[si], false, b[sj], (short)0, c[si][sj],
                        false, false);
        }

        // C/D layout: lane, VGPR v -> (M = v + 8h, N = lane&15)
        const bool  diagM = (ti == tj);
        const float wgt   = diagM ? 1.0f : 2.0f;  // symmetry weight

        float invj[4];
        #pragma unroll
        for (int sj = 0; sj < 4; ++sj) invj[sj] = inv[mj0 + sj * 16 + n];

        #pragma unroll
        for (int si = 0; si < 4; ++si) {
            #pragma unroll
            for (int v = 0; v < 8; ++v) {
                const int   i    = mi0 + si * 16 + v + 8 * h;
                const float invi = inv[i];
                #pragma unroll
                for (int sj = 0; sj < 4; ++sj) {
                    const int j = mj0 + sj * 16 + n;
                    float s = c[si][sj][v] * invi * invj[sj]; // cosine sim
                    float e = __expf(2.0f * s);               // exp(s/tau)
                    float w = (diagM && (i == j)) ? 0.0f : wgt;
                    tot += w * e;
                    const int d = j - i;                      // |d|<64 on diagM
                    if (!diagM && (d == 2048 || d == 4096 || d == 6144))
                        nom += w * e;                         // w==2: (i,j)+(j,i)
                }
            }
        }
    }

    // block reduction, then fp64 atomics
    __shared__ float st[256], sn[256];
    st[threadIdx.x] = tot;
    sn[threadIdx.x] = nom;
    __syncthreads();
    for (int off = 128; off > 0; off >>= 1) {
        if (threadIdx.x < off) {
            st[threadIdx.x] += st[threadIdx.x + off];
            sn[threadIdx.x] += sn[threadIdx.x + off];
        }
        __syncthreads();
    }
    if (threadIdx.x == 0) {
        atomicAdd(&acc[0], (double)st[0]);
        atomicAdd(&acc[1], (double)sn[0]);
    }
}

// ---- Kernel 3: finalize ----------------------------------------------------
__global__ void ntx_finalize_kernel(const double* __restrict__ acc,
                                    float* __restrict__ out)
{
    const double total = acc[0];         // sum over i!=j of exp(sim/tau)
    const double nomv  = acc[1];
    const double denom = total - nomv;
    out[0] = (float)(-log(nomv / denom) / (double)NROWS);
}

// ---------------------------------------------------------------------------
extern "C" void kernel_launch(void* const* d_in, const int* in_sizes, int n_in,
                              void* d_out, int out_size, void* d_ws, size_t ws_size,
                              hipStream_t stream)
{
    (void)in_sizes; (void)n_in; (void)out_size; (void)ws_size;

    const float* ei = (const float*)d_in[0];     // emb_i [4096, 512] f32
    const float* ej = (const float*)d_in[1];     // emb_j [4096, 512] f32
    float* out = (float*)d_out;                  // scalar loss, f32

    // ws layout: [0,16): two fp64 accumulators; [64, 64+32KB): inv norms
    double* acc = (double*)d_ws;
    float*  inv = (float*)((char*)d_ws + 64);

    hipMemsetAsync(d_ws, 0, 64, stream);         // zero accumulators (graph-safe)

    ntx_rownorm_kernel<<<NROWS, 256, 0, stream>>>(ei, ej, inv);
    // 1032 blocks * 8 waves = 8256 waves: exactly one 64x64 macro per wave
    ntx_sim_kernel<<<1032, 256, 0, stream>>>(ei, ej, inv, acc);
    ntx_finalize_kernel<<<1, 1, 0, stream>>>(acc, out);
}